// TensorProductLinearGate_44203803411115
// MI455X (gfx1250) — compile-verified
//
#include <hip/hip_runtime.h>
#include <math.h>

typedef __attribute__((ext_vector_type(2))) float v2f;
typedef __attribute__((ext_vector_type(4))) float f4;
typedef __attribute__((ext_vector_type(8))) float v8f;

#define INV_SQRT_FAN 0.08838834764831845f   // 1/sqrt(128)
#define INV_SQRT3    0.5773502691896258f    // 1/sqrt(3)
#define SQRT_2_PI    0.7978845608028654f    // sqrt(2/pi)

// LDS layout (floats), reused across phases:
//   phase A: staged x block: 32 rows, stride 260  -> [0, 8320)
//   phase B: s  : 2 groups x 16 rows x stride 132 -> [0, 4224)
//            v  : 2 groups x 16 rows x stride 196 -> [4224, 10496)
#define LDS_FLOATS 10496
#define SX_STRIDE 260
#define SS_STRIDE 132
#define SV_STRIDE 196
#define SV_OFF 4224

__device__ __forceinline__ float gelu_tanh(float u) {
    float u3 = u * u * u;
    float t  = tanhf(SQRT_2_PI * (u + 0.044715f * u3));
    return 0.5f * u * (1.0f + t);
}

__global__ __launch_bounds__(256) void tplg_kernel(
    const float* __restrict__ x, const float* __restrict__ y,
    const float* __restrict__ W0, const float* __restrict__ b0,
    const float* __restrict__ W1, float* __restrict__ out)
{
    __shared__ float lds[LDS_FLOATS];

    const int tid = threadIdx.x;
    const int rb  = blockIdx.x * 32;          // first row of this block

    // ---- stage x: 32 rows x 256 f32, contiguous in global, padded in LDS ----
    {
        const f4* xg = (const f4*)(x + (size_t)rb * 256);
        #pragma unroll
        for (int it = 0; it < 8; ++it) {
            int f4i = it * 256 + tid;         // 0..2047
            int row = f4i >> 6;               // f4i / 64  (64 f4 per row)
            int col = (f4i & 63) << 2;
            f4 v = xg[f4i];
            float* d = &lds[row * SX_STRIDE + col];
            d[0] = v.x; d[1] = v.y; d[2] = v.z; d[3] = v.w;
        }
    }
    __syncthreads();

    const int wave = tid >> 5, lane = tid & 31;
    const int g = wave >> 2, q = wave & 3;    // row-group, column-wave
    const int r = lane & 15, h = lane >> 4;   // lane row, lane half

    // per-lane y for row rb + g*16 + r : (y0, y1x, y1y, y1z)
    f4 yv = ((const f4*)y)[rb + g * 16 + r];
    const float y0v = yv.x, y1x = yv.y, y1y = yv.z, y1z = yv.w;

    v8f accS0 = {}, accS1 = {}, accV0 = {}, accV1 = {}, accV2 = {};

    const float* sx  = &lds[(g * 16 + r) * SX_STRIDE];
    const int colS = 32 * q + r;              // W0 / s column, tile t=0 (t=1: +16)
    const int colV = 16 * q + r;              // W1 / v column

    // ---- K-steps 0..15 : tp columns c < 64 (x0 path) ----
    #pragma unroll 4
    for (int k = 0; k < 16; ++k) {
        const int c0 = 4 * k + 2 * h;         // K = c0, c0+1 this lane
        v2f xa = *(const v2f*)(sx + c0);      // x0[r][c0], x0[r][c0+1]
        v2f a0  = xa * y0v;                   // tp0
        v2f a10 = xa * y1x;                   // tp1_x
        v2f a11 = xa * y1y;                   // tp1_y
        v2f a12 = xa * y1z;                   // tp1_z

        const float* w0r = W0 + c0 * 128;
        const float* w1r = W1 + c0 * 64;
        v2f bS0 = (v2f){ w0r[colS],      w0r[128 + colS]      };
        v2f bS1 = (v2f){ w0r[colS + 16], w0r[128 + colS + 16] };
        v2f bV  = (v2f){ w1r[colV],      w1r[64 + colV]       };

        accS0 = __builtin_amdgcn_wmma_f32_16x16x4_f32(false, a0,  false, bS0, (short)0, accS0, false, false);
        accS1 = __builtin_amdgcn_wmma_f32_16x16x4_f32(false, a0,  false, bS1, (short)0, accS1, false, false);
        accV0 = __builtin_amdgcn_wmma_f32_16x16x4_f32(false, a10, false, bV,  (short)0, accV0, false, false);
        accV1 = __builtin_amdgcn_wmma_f32_16x16x4_f32(false, a11, false, bV,  (short)0, accV1, false, false);
        accV2 = __builtin_amdgcn_wmma_f32_16x16x4_f32(false, a12, false, bV,  (short)0, accV2, false, false);
    }

    // ---- K-steps 16..31 : tp columns c >= 64 (x1 path) ----
    #pragma unroll 4
    for (int kk = 0; kk < 16; ++kk) {
        const int cc = 4 * kk + 2 * h;        // x1 channel cc, cc+1
        const int c0 = 64 + cc;               // tp column
        const float* p = sx + 64 + 3 * cc;    // 6 consecutive floats
        float e0 = p[0], e1 = p[1], e2 = p[2], e3 = p[3], e4 = p[4], e5 = p[5];
        v2f a0  = (v2f){ (e0 * y1x + e1 * y1y + e2 * y1z) * INV_SQRT3,
                         (e3 * y1x + e4 * y1y + e5 * y1z) * INV_SQRT3 };
        v2f a10 = (v2f){ e0 * y0v, e3 * y0v };
        v2f a11 = (v2f){ e1 * y0v, e4 * y0v };
        v2f a12 = (v2f){ e2 * y0v, e5 * y0v };

        const float* w0r = W0 + c0 * 128;
        const float* w1r = W1 + c0 * 64;
        v2f bS0 = (v2f){ w0r[colS],      w0r[128 + colS]      };
        v2f bS1 = (v2f){ w0r[colS + 16], w0r[128 + colS + 16] };
        v2f bV  = (v2f){ w1r[colV],      w1r[64 + colV]       };

        accS0 = __builtin_amdgcn_wmma_f32_16x16x4_f32(false, a0,  false, bS0, (short)0, accS0, false, false);
        accS1 = __builtin_amdgcn_wmma_f32_16x16x4_f32(false, a0,  false, bS1, (short)0, accS1, false, false);
        accV0 = __builtin_amdgcn_wmma_f32_16x16x4_f32(false, a10, false, bV,  (short)0, accV0, false, false);
        accV1 = __builtin_amdgcn_wmma_f32_16x16x4_f32(false, a11, false, bV,  (short)0, accV1, false, false);
        accV2 = __builtin_amdgcn_wmma_f32_16x16x4_f32(false, a12, false, bV,  (short)0, accV2, false, false);
    }

    __syncthreads();   // everyone done reading staged x before LDS reuse

    // ---- scatter s (scaled + bias) and v (scaled) tiles into LDS ----
    const float bb0 = b0[colS], bb1 = b0[colS + 16];
    #pragma unroll
    for (int vi = 0; vi < 8; ++vi) {
        int row = vi + 8 * h;                 // C/D layout: M = vgpr + 8*half
        float* sS = &lds[g * (16 * SS_STRIDE) + row * SS_STRIDE];
        sS[colS]      = accS0[vi] * INV_SQRT_FAN + bb0;
        sS[colS + 16] = accS1[vi] * INV_SQRT_FAN + bb1;
        float* sV = &lds[SV_OFF + g * (16 * SV_STRIDE) + row * SV_STRIDE];
        sV[colV * 3 + 0] = accV0[vi] * INV_SQRT_FAN;
        sV[colV * 3 + 1] = accV1[vi] * INV_SQRT_FAN;
        sV[colV * 3 + 2] = accV2[vi] * INV_SQRT_FAN;
    }
    __syncthreads();

    // ---- epilogue: gates/gelu, coalesced b128 stores ----
    {
        const int ar = tid >> 3;              // 0..31 local row
        const int chunk = tid & 7;            // 32-col chunk
        const int g2 = ar >> 4, r2 = ar & 15;
        const float* sS = &lds[g2 * (16 * SS_STRIDE) + r2 * SS_STRIDE];
        const float* sV = &lds[SV_OFF + g2 * (16 * SV_STRIDE) + r2 * SV_STRIDE];
        float* og = out + (size_t)(rb + ar) * 256 + chunk * 32;

        #pragma unroll
        for (int e = 0; e < 8; ++e) {
            int base = chunk * 32 + e * 4;
            float t[4];
            #pragma unroll
            for (int j = 0; j < 4; ++j) {
                int col = base + j;
                if (col < 64) {
                    t[j] = gelu_tanh(sS[64 + col]);     // scalars = gelu(s[:,64:])
                } else {
                    int m = col - 64;                   // m = o*3 + i
                    int o = m / 3;
                    float gate = 1.0f / (1.0f + expf(-sS[o]));
                    t[j] = gate * sV[m];
                }
            }
            f4 v = (f4){ t[0], t[1], t[2], t[3] };
            *(f4*)(og + e * 4) = v;
        }
    }
}

extern "C" void kernel_launch(void* const* d_in, const int* in_sizes, int n_in,
                              void* d_out, int out_size, void* d_ws, size_t ws_size,
                              hipStream_t stream) {
    const float* x  = (const float*)d_in[0];
    const float* y  = (const float*)d_in[1];
    const float* W0 = (const float*)d_in[2];
    const float* b0 = (const float*)d_in[3];
    const float* W1 = (const float*)d_in[4];
    float* out = (float*)d_out;

    const int n = in_sizes[0] / 256;          // rows (x is N x 256)
    const int blocks = n / 32;                // 32 rows per workgroup
    tplg_kernel<<<blocks, 256, 0, stream>>>(x, y, W0, b0, W1, out);
}